// BERTClassifier_35124242546782
// MI455X (gfx1250) — compile-verified
//
#include <hip/hip_runtime.h>
#include <math.h>

#define B_ 32
#define L_ 512
#define H_ 768
#define P_ 5
#define K_ 5
#define C_ 4
#define N_ (B_ * P_)   // 160

#define LOG2PI_      1.8378770664093453f
#define INV_SQRT2PI_ 0.3989422804014327f

typedef __attribute__((ext_vector_type(2))) float v2f;
typedef __attribute__((ext_vector_type(8))) float v8f;

__device__ __forceinline__ float wave_reduce_sum(float v) {
#pragma unroll
  for (int off = 16; off > 0; off >>= 1)
    v += __shfl_xor(v, off, 32);
  return v;
}

// ---------------------------------------------------------------------------
// K1: prototype norms + normalized prototypes.  One wave per prototype.
// ---------------------------------------------------------------------------
__global__ void proto_norm_kernel(const float* __restrict__ proto,
                                  float* __restrict__ p_n,
                                  float* __restrict__ pnorm) {
  int wave = threadIdx.x >> 5;
  int lane = threadIdx.x & 31;
  if (wave >= P_) return;
  float s = 0.f;
  for (int h = lane; h < H_; h += 32) {
    float x = proto[wave * H_ + h];
    s += x * x;
  }
  s = wave_reduce_sum(s);
  float nrm = sqrtf(s);
  if (lane == 0) pnorm[wave] = nrm;
  float inv = 1.0f / fmaxf(nrm, 1e-12f);
  for (int h = lane; h < H_; h += 32)
    p_n[wave * H_ + h] = proto[wave * H_ + h] * inv;
}

// ---------------------------------------------------------------------------
// K2: cosine similarity cs[(b*P+p), l].  One wave per (b,l) row; padded_emb
// is read exactly once, coalesced along H.
// ---------------------------------------------------------------------------
__global__ void cos_sim_kernel(const float* __restrict__ emb,
                               const float* __restrict__ p_n,
                               float* __restrict__ cs) {
  int wave = threadIdx.x >> 5;
  int lane = threadIdx.x & 31;
  int row  = blockIdx.x * (blockDim.x >> 5) + wave;
  if (row >= B_ * L_) return;
  int b = row / L_, l = row % L_;
  const float* a = emb + (size_t)row * H_;
  float nrm = 0.f, d0 = 0.f, d1 = 0.f, d2 = 0.f, d3 = 0.f, d4 = 0.f;
  for (int h = lane; h < H_; h += 32) {
    float x = a[h];
    nrm += x * x;
    d0 += x * p_n[0 * H_ + h];
    d1 += x * p_n[1 * H_ + h];
    d2 += x * p_n[2 * H_ + h];
    d3 += x * p_n[3 * H_ + h];
    d4 += x * p_n[4 * H_ + h];
  }
  nrm = wave_reduce_sum(nrm);
  d0 = wave_reduce_sum(d0);
  d1 = wave_reduce_sum(d1);
  d2 = wave_reduce_sum(d2);
  d3 = wave_reduce_sum(d3);
  d4 = wave_reduce_sum(d4);
  if (lane == 0) {
    float inv = 1.0f / fmaxf(sqrtf(nrm), 1e-12f);
    cs[((size_t)(b * P_ + 0)) * L_ + l] = d0 * inv;
    cs[((size_t)(b * P_ + 1)) * L_ + l] = d1 * inv;
    cs[((size_t)(b * P_ + 2)) * L_ + l] = d2 * inv;
    cs[((size_t)(b * P_ + 3)) * L_ + l] = d3 * inv;
    cs[((size_t)(b * P_ + 4)) * L_ + l] = d4 * inv;
  }
}

// ---------------------------------------------------------------------------
// K3: h = tanh(cs @ W1 + b1) with V_WMMA_F32_16X16X4_F32.
// One wave per 16x16 output tile; K=512 in steps of 4.
// ISA f32 fragment layouts: r = lane&15, kh = lane>>4;
//   A frag: a[j] = A[m0+r][k0 + j + 2*kh]
//   B frag: b[j] = B[k0 + j + 2*kh][n0+r]
//   C/D:    acc[i] = C[m0 + i + 8*kh][n0+r]
// ---------------------------------------------------------------------------
__global__ void gemm_tanh_wmma(const float* __restrict__ A,   // [N_, L_]
                               const float* __restrict__ W1,  // [L_, L_]
                               const float* __restrict__ b1,  // [L_]
                               float* __restrict__ Hout) {    // [N_, L_]
  int ln = threadIdx.x;
  int r  = ln & 15;
  int kh = ln >> 4;
  int n0 = blockIdx.x * 16;
  int m0 = blockIdx.y * 16;
  const float* arow = A + (size_t)(m0 + r) * L_;
  v8f acc = {};
  for (int k0 = 0; k0 < L_; k0 += 4) {
    int ka = k0 + 2 * kh;
    v2f af, bf;
    af[0] = arow[ka + 0];
    af[1] = arow[ka + 1];
    bf[0] = W1[(size_t)(ka + 0) * L_ + n0 + r];
    bf[1] = W1[(size_t)(ka + 1) * L_ + n0 + r];
    acc = __builtin_amdgcn_wmma_f32_16x16x4_f32(
        /*neg_a=*/false, af, /*neg_b=*/false, bf,
        /*c_mod=*/(short)0, acc, /*reuse_a=*/false, /*reuse_b=*/false);
  }
  int n = n0 + r;
  float bias = b1[n];
#pragma unroll
  for (int i = 0; i < 8; ++i) {
    int m = m0 + i + 8 * kh;
    Hout[(size_t)m * L_ + n] = tanhf(acc[i] + bias);
  }
}

// ---------------------------------------------------------------------------
// K4: MDN heads + top-K labels + NLL row sums + density mask + denom.
// One wave per row n in [0, 160).
// ---------------------------------------------------------------------------
__global__ void mdn_kernel(const float* __restrict__ Hmat,  // [N_, L_]
                           const float* __restrict__ cs,    // [N_, L_]
                           const float* __restrict__ Wpi, const float* __restrict__ bpi,
                           const float* __restrict__ Wmu, const float* __restrict__ bmu,
                           const float* __restrict__ Wsig, const float* __restrict__ bsig,
                           float* __restrict__ mask,        // [N_, L_]
                           float* __restrict__ denom,       // [N_]
                           float* __restrict__ ll_sum) {    // [N_]
  int n = blockIdx.x;
  int lane = threadIdx.x;
  const float* hrow = Hmat + (size_t)n * L_;

  float acc[3 * K_];
#pragma unroll
  for (int i = 0; i < 3 * K_; ++i) acc[i] = 0.f;
  for (int l = lane; l < L_; l += 32) {
    float hv = hrow[l];
#pragma unroll
    for (int k = 0; k < K_; ++k) {
      acc[k]          += hv * Wpi [l * K_ + k];
      acc[K_ + k]     += hv * Wmu [l * K_ + k];
      acc[2 * K_ + k] += hv * Wsig[l * K_ + k];
    }
  }
#pragma unroll
  for (int i = 0; i < 3 * K_; ++i) acc[i] = wave_reduce_sum(acc[i]);

  float rawpi[K_], mu[K_], ls[K_], sig[K_], logpi[K_], pi[K_];
  float mx = -INFINITY;
#pragma unroll
  for (int k = 0; k < K_; ++k) {
    rawpi[k] = acc[k] + bpi[k];
    mu[k]    = acc[K_ + k] + bmu[k];
    ls[k]    = acc[2 * K_ + k] + bsig[k];
    sig[k]   = expf(ls[k]);
    mx = fmaxf(mx, rawpi[k]);
  }
  float se = 0.f;
#pragma unroll
  for (int k = 0; k < K_; ++k) se += expf(rawpi[k] - mx);
  float lse = mx + logf(se);
#pragma unroll
  for (int k = 0; k < K_; ++k) {
    logpi[k] = rawpi[k] - lse;
    pi[k]    = expf(logpi[k]);
  }

  // ---- top-K (K=5) positions of cs row, jax tie-break (lowest index) ----
  const float* crow = cs + (size_t)n * L_;
  float myv[L_ / 32];
#pragma unroll
  for (int i = 0; i < L_ / 32; ++i) myv[i] = crow[lane + 32 * i];
  int picked[K_];
  float y[K_];
#pragma unroll
  for (int rnd = 0; rnd < K_; ++rnd) {
    float bv = -INFINITY;
    int   bi = L_;  // sentinel
#pragma unroll
    for (int i = 0; i < L_ / 32; ++i) {
      int l = lane + 32 * i;
      bool skip = false;
      for (int q = 0; q < rnd; ++q)
        if (picked[q] == l) skip = true;
      float v = myv[i];
      if (!skip && (v > bv || (v == bv && l < bi))) { bv = v; bi = l; }
    }
    for (int off = 16; off > 0; off >>= 1) {
      float ov = __shfl_xor(bv, off, 32);
      int   oi = __shfl_xor(bi, off, 32);
      if (ov > bv || (ov == bv && oi < bi)) { bv = ov; bi = oi; }
    }
    picked[rnd] = bi;
    y[rnd] = (float)bi;
  }

  // ---- NLL row sum: sum_j logsumexp_k(log_pi + comp_logp) ----
  float llsum = 0.f;
#pragma unroll
  for (int j = 0; j < K_; ++j) {
    float t[K_], tmx = -INFINITY;
#pragma unroll
    for (int k = 0; k < K_; ++k) {
      float z = (y[j] - mu[k]) / sig[k];
      t[k] = logpi[k] - 0.5f * z * z - ls[k] - 0.5f * LOG2PI_;
      tmx = fmaxf(tmx, t[k]);
    }
    float s = 0.f;
#pragma unroll
    for (int k = 0; k < K_; ++k) s += expf(t[k] - tmx);
    llsum += tmx + logf(s);
  }
  if (lane == 0) ll_sum[n] = llsum;

  // ---- density mask over token positions + denom ----
  float dsum = 0.f;
  for (int l = lane; l < L_; l += 32) {
    float d = 0.f;
#pragma unroll
    for (int k = 0; k < K_; ++k) {
      float zt = ((float)l - mu[k]) / sig[k];
      d += pi[k] * expf(-0.5f * zt * zt) / sig[k] * INV_SQRT2PI_;
    }
    mask[(size_t)n * L_ + l] = d;
    dsum += d;
  }
  dsum = wave_reduce_sum(dsum);
  if (lane == 0) denom[n] = fmaxf(dsum, 1e-9f);
}

// ---------------------------------------------------------------------------
// K5: masked mean pooling Z[b,p,h].  Mask tile (10 KB) staged into LDS via
// GLOBAL_LOAD_ASYNC_TO_LDS_B128 (ASYNCcnt path, no VGPR staging), then
// hidden_states is streamed exactly once, coalesced along H.
// ---------------------------------------------------------------------------
__global__ void pool_kernel(const float* __restrict__ hs,    // [B,L,H]
                            const float* __restrict__ mask,  // [B,P,L]
                            const float* __restrict__ denom, // [B,P]
                            float* __restrict__ Z) {         // [B,P,H]
  __shared__ float sm[P_][L_];
  int b = blockIdx.y;
  int tid = threadIdx.x;

  // Async stage mask[b,:,:] (P_*L_ = 2560 floats) into LDS as 640 x b128.
  // Generic pointers to __shared__ carry the LDS byte offset in their low
  // 32 bits (hardware truncates addr[31:0] in the LDS aperture).
  {
    const float* src = mask + (size_t)b * (P_ * L_);
    unsigned lds_base = (unsigned)(size_t)(&sm[0][0]);
    for (int q = tid; q < (P_ * L_) / 4; q += 256) {
      unsigned ldsa = lds_base + (unsigned)q * 16u;
      const float* g = src + (size_t)q * 4;
      asm volatile("global_load_async_to_lds_b128 %0, %1, off"
                   :: "v"(ldsa), "v"(g)
                   : "memory");
    }
    asm volatile("s_wait_asynccnt 0x0" ::: "memory");
  }
  __syncthreads();

  int h = blockIdx.x * 256 + tid;
  const float* hb = hs + (size_t)b * L_ * H_;
  float acc[P_] = {0.f, 0.f, 0.f, 0.f, 0.f};
  for (int l = 0; l < L_; ++l) {
    float hv = hb[(size_t)l * H_ + h];
#pragma unroll
    for (int p = 0; p < P_; ++p) acc[p] += hv * sm[p][l];
  }
#pragma unroll
  for (int p = 0; p < P_; ++p) {
    float dn = denom[b * P_ + p];
    Z[((size_t)(b * P_ + p)) * H_ + h] = acc[p] / ((float)L_ * dn);
  }
}

// ---------------------------------------------------------------------------
// K6: cosine similarity of Z vs (unnormalized) prototypes. One wave per (b,p).
// ---------------------------------------------------------------------------
__global__ void sim_kernel(const float* __restrict__ Z,
                           const float* __restrict__ proto,
                           const float* __restrict__ pnorm,
                           float* __restrict__ sim) {
  int n = blockIdx.x;
  int p = n % P_;
  int lane = threadIdx.x;
  float dot = 0.f, zn = 0.f;
  for (int h = lane; h < H_; h += 32) {
    float zv = Z[(size_t)n * H_ + h];
    dot += zv * proto[p * H_ + h];
    zn  += zv * zv;
  }
  dot = wave_reduce_sum(dot);
  zn  = wave_reduce_sum(zn);
  if (lane == 0) {
    float den = sqrtf(zn) * pnorm[p];
    sim[n] = dot / fmaxf(den, 1e-6f);
  }
}

// ---------------------------------------------------------------------------
// K7: logits, loss_mu, diversity.  Single block; deterministic reductions.
// out layout: [0..127]=logits, [128]=loss_mu, [129]=diversity_loss
// ---------------------------------------------------------------------------
__global__ void finalize_kernel(const float* __restrict__ sim,     // [B,P]
                                const float* __restrict__ fcW,     // [P,C]
                                const float* __restrict__ fcb,     // [C]
                                const float* __restrict__ ll_sum,  // [N]
                                const float* __restrict__ proto,   // [P,H]
                                float* __restrict__ out) {
  __shared__ float red[256];
  int tid = threadIdx.x;

  if (tid < B_ * C_) {
    int b = tid / C_, c = tid % C_;
    float lg = fcb[c];
#pragma unroll
    for (int p = 0; p < P_; ++p) lg += sim[b * P_ + p] * fcW[p * C_ + c];
    out[tid] = lg;
  }

  red[tid] = (tid < N_) ? ll_sum[tid] : 0.f;
  __syncthreads();
  for (int s = 128; s > 0; s >>= 1) {
    if (tid < s) red[tid] += red[tid + s];
    __syncthreads();
  }
  if (tid == 0) out[B_ * C_] = -red[0] / (float)(N_ * K_);
  __syncthreads();

  float dv = 0.f;
  if (tid < (P_ * (P_ - 1)) / 2) {
    int i = 0, j = 0, cnt = 0;
    for (int a = 0; a < P_; ++a)
      for (int bb = a + 1; bb < P_; ++bb) {
        if (cnt == tid) { i = a; j = bb; }
        ++cnt;
      }
    float d2 = 0.f;
    for (int h = 0; h < H_; ++h) {
      float d = proto[i * H_ + h] - proto[j * H_ + h];
      d2 += d * d;
    }
    float D = sqrtf(fmaxf(d2, 1e-12f));
    dv = fmaxf(0.5f - D, 0.f);
  }
  red[tid] = dv;
  __syncthreads();
  for (int s = 128; s > 0; s >>= 1) {
    if (tid < s) red[tid] += red[tid + s];
    __syncthreads();
  }
  if (tid == 0) out[B_ * C_ + 1] = red[0];
}

// ---------------------------------------------------------------------------
extern "C" void kernel_launch(void* const* d_in, const int* in_sizes, int n_in,
                              void* d_out, int out_size, void* d_ws, size_t ws_size,
                              hipStream_t stream) {
  (void)in_sizes; (void)n_in; (void)out_size; (void)ws_size;
  const float* emb   = (const float*)d_in[0];
  const float* hs    = (const float*)d_in[1];
  const float* proto = (const float*)d_in[2];
  const float* W1    = (const float*)d_in[3];
  const float* b1    = (const float*)d_in[4];
  const float* Wpi   = (const float*)d_in[5];
  const float* bpi   = (const float*)d_in[6];
  const float* Wmu   = (const float*)d_in[7];
  const float* bmu   = (const float*)d_in[8];
  const float* Wsig  = (const float*)d_in[9];
  const float* bsig  = (const float*)d_in[10];
  const float* fcW   = (const float*)d_in[11];
  const float* fcb   = (const float*)d_in[12];
  float* out = (float*)d_out;

  float* ws    = (float*)d_ws;
  float* p_n   = ws;                       // P*H       = 3840
  float* pnorm = p_n + (size_t)P_ * H_;    // 5 (pad 8)
  float* cs    = pnorm + 8;                // N*L       = 81920
  float* Hm    = cs + (size_t)N_ * L_;     // N*L
  float* mk    = Hm + (size_t)N_ * L_;     // N*L
  float* denom = mk + (size_t)N_ * L_;     // N
  float* llsum = denom + N_;               // N
  float* Z     = llsum + N_;               // N*H       = 122880
  float* sim   = Z + (size_t)N_ * H_;      // N

  proto_norm_kernel<<<1, 256, 0, stream>>>(proto, p_n, pnorm);
  cos_sim_kernel<<<(B_ * L_) / 8, 256, 0, stream>>>(emb, p_n, cs);
  gemm_tanh_wmma<<<dim3(L_ / 16, N_ / 16), 32, 0, stream>>>(cs, W1, b1, Hm);
  mdn_kernel<<<N_, 32, 0, stream>>>(Hm, cs, Wpi, bpi, Wmu, bmu, Wsig, bsig,
                                    mk, denom, llsum);
  pool_kernel<<<dim3(H_ / 256, B_), 256, 0, stream>>>(hs, mk, denom, Z);
  sim_kernel<<<N_, 32, 0, stream>>>(Z, proto, pnorm, sim);
  finalize_kernel<<<1, 256, 0, stream>>>(sim, fcW, fcb, llsum, proto, out);
}